// KAN_HS_GAL_71244917506578
// MI455X (gfx1250) — compile-verified
//
#include <hip/hip_runtime.h>
#include <math.h>

typedef __attribute__((ext_vector_type(2))) float v2f;
typedef __attribute__((ext_vector_type(8))) float v8f;

// ---------------- constants ----------------
#define KPF 20                 // GRID_SIZE + ORDER basis funcs per feature
#define GSTEP (3.125f/24.0f)   // uniform grid spacing: linspace(-1.5, 1.625, 25)
#define GMIN  (-1.5f)

__device__ __forceinline__ v8f wmma4(v2f a, v2f b, v8f c) {
    // D = A(16x4,f32) * B(4x16,f32) + C  -- fp32 WMMA, keeps reference precision
    return __builtin_amdgcn_wmma_f32_16x16x4_f32(false, a, false, b, (short)0, c, false, false);
}

// Dense de Boor recursion, order 4, 20 outputs, mirrors reference bspline_basis.
// Writes lds[(fl*20+t)*16 + r] for t in [0,20).
__device__ __forceinline__ void basis20(float x, float* lds, int r, int fl) {
    float xc = fminf(fmaxf(x, -1.0f), 1.0f);
    float b[24];
#pragma unroll
    for (int i = 0; i < 24; ++i) {
        float gi  = GMIN + GSTEP * i;
        float gi1 = gi + GSTEP;
        b[i] = (xc >= gi && xc < gi1) ? 1.0f : 0.0f;
    }
#pragma unroll
    for (int k = 1; k <= 4; ++k) {
        float inv = 1.0f / (k * GSTEP);
#pragma unroll 23
        for (int i = 0; i < 24 - k; ++i) {
            float gi   = GMIN + GSTEP * i;
            float gik1 = GMIN + GSTEP * (i + k + 1);
            b[i] = ((xc - gi) * b[i] + (gik1 - xc) * b[i + 1]) * inv;
        }
    }
#pragma unroll
    for (int t = 0; t < KPF; ++t) lds[(fl * KPF + t) * 16 + r] = b[t];
}

// GEMM over one K-chunk. Weights are pair-interleaved: element k, col o lives at
// WcP[(k>>1)*(FOUT*2) + o*2 + (k&1)]  ->  lane loads one contiguous b64.
template <int KLOC, int FOUT>
__device__ __forceinline__ void gemm_chunk(const float* __restrict__ BpPair,
                                           const float* lds, v8f& acc,
                                           int m, int kh) {
#pragma unroll 4
    for (int kb = 0; kb < KLOC; kb += 4) {
        int k0 = kb + (kh << 1);
        v2f a;
        a[0] = lds[(k0 + 0) * 16 + m];
        a[1] = lds[(k0 + 1) * 16 + m];
        v2f b = *(const v2f*)(BpPair + (size_t)(k0 >> 1) * (FOUT * 2));
        acc = wmma4(a, b, acc);
    }
}

// ------------- generic KAN 16-row tile GEMM (blockDim must be 128 = 4 waves) -------------
// Contraction layout (matches k_prep): k = i*20+t (spline), FIN*20+i (neg), FIN*21+i (pos).
template <int FIN, int FOUT, class In>
__device__ v8f kan_gemm16(const In& in, const float* __restrict__ WcP, float* lds) {
    const int tid = threadIdx.x;
    const int wave = tid >> 5, lane = tid & 31;
    const int m = lane & 15, kh = lane >> 4;
    const bool active = (wave << 4) < FOUT;
    v8f acc = {};
    constexpr int NF = FIN >> 4;     // spline chunks of 16 features (K=320 each)
    int kGlob = 0;
#pragma unroll 1
    for (int c = 0; c < NF; ++c) {
        const float* wbase = WcP + (size_t)(kGlob >> 1) * (FOUT * 2);
        // prefetch this chunk's weights into cache while VALU computes the basis
        for (int off = tid * 128; off < 320 * FOUT * 4; off += 128 * 128)
            __builtin_prefetch((const char*)wbase + off, 0, 1);
        for (int item = tid; item < 256; item += 128) {
            int fl = item >> 4, r = item & 15;
            basis20(in(r, (c << 4) + fl), lds, r, fl);
        }
        __syncthreads();
        if (active)
            gemm_chunk<16 * KPF, FOUT>(wbase + ((wave << 4) + m) * 2, lds, acc, m, kh);
        __syncthreads();
        kGlob += 16 * KPF;
    }
    // base chunk: neg | pos  (K = 2*FIN)
    {
        const float* wbase = WcP + (size_t)(kGlob >> 1) * (FOUT * 2);
        for (int off = tid * 128; off < 2 * FIN * FOUT * 4; off += 128 * 128)
            __builtin_prefetch((const char*)wbase + off, 0, 1);
        for (int item = tid; item < 16 * FIN; item += 128) {
            int fl = item >> 4, r = item & 15;
            float xc = fminf(fmaxf(in(r, fl), -1.0f), 1.0f);
            float pos = fmaxf(xc, 0.0f);
            lds[fl * 16 + r] = xc - pos;          // neg
            lds[(FIN + fl) * 16 + r] = pos;       // pos
        }
        __syncthreads();
        if (active)
            gemm_chunk<2 * FIN, FOUT>(wbase + ((wave << 4) + m) * 2, lds, acc, m, kh);
        __syncthreads();
    }
    return acc;
}

// ---------------- input functors ----------------
struct InPlain {
    const float* X; int ld; int row0; int R;
    __device__ float operator()(int r, int f) const {
        int row = row0 + r;
        return (row < R) ? X[(size_t)row * ld + f] : 0.0f;
    }
};
struct InPair {  // prod[b,i,j,f] = h_st[b,i,f]*h_st[b,j,f]; row = ((b*64)+i)*64+j
    const float* Hst; int row0;
    __device__ float operator()(int r, int f) const {
        int row = row0 + r;
        int b = row >> 12, i = (row >> 6) & 63, j = row & 63;
        return Hst[(((b << 6) + i) << 6) + f] * Hst[(((b << 6) + j) << 6) + f];
    }
};
struct InStack { // h_st * pad(S); row = b*64+n
    const float* Hst; const float* S; int row0;
    __device__ float operator()(int r, int f) const {
        int row = row0 + r;
        int b = row >> 6;
        float s = (f < 32) ? S[(b << 5) + f] : 0.0f;
        return Hst[((size_t)row << 6) + f] * s;
    }
};

// ---------------- kernels ----------------

// Build combined weight matrix, pair-interleaved over k:
// WcP[(k>>1)*(fout*2) + o*2 + (k&1)], k in [0, 22*fin)
__global__ void k_prep(const float* __restrict__ prelu, const float* __restrict__ base,
                       const float* __restrict__ spl, const float* __restrict__ coef,
                       float* __restrict__ WcP, int fin, int fout) {
    int total = fin * 22 * fout;
    for (int idx = blockIdx.x * blockDim.x + threadIdx.x; idx < total;
         idx += gridDim.x * blockDim.x) {
        int o = idx % fout;
        int k = idx / fout;
        float w;
        if (k < fin * KPF) {
            int i = k / KPF, t = k % KPF;
            w = spl[o * fin + i] * coef[(o * fin + i) * KPF + t];
        } else if (k < fin * (KPF + 1)) {
            int i = k - fin * KPF;
            w = base[o * fin + i] * prelu[o * fin + i];   // neg weights
        } else {
            int i = k - fin * (KPF + 1);
            w = base[o * fin + i];                         // pos weights
        }
        WcP[(size_t)(k >> 1) * (fout * 2) + o * 2 + (k & 1)] = w;
    }
}

// Generic KAN layer: Y[row, colOff+o] (+)= kan(X)[row,o]
template <int FIN, int FOUT>
__global__ void __launch_bounds__(128)
k_kan_plain(const float* __restrict__ X, int ldx, const float* __restrict__ WcP,
            float* __restrict__ Y, int ldy, int colOff, int addFlag, int R) {
    __shared__ float lds[5120];
    int row0 = blockIdx.x << 4;
    InPlain in{X, ldx, row0, R};
    v8f acc = kan_gemm16<FIN, FOUT>(in, WcP, lds);
    int tid = threadIdx.x, wave = tid >> 5, lane = tid & 31;
    int kh = lane >> 4, n = lane & 15;
    if ((wave << 4) < FOUT) {
#pragma unroll
        for (int r = 0; r < 8; ++r) {
            int row = row0 + r + (kh << 3);
            if (row < R) {
                float* p = &Y[(size_t)row * ldy + colOff + (wave << 4) + n];
                float v = acc[r];
                if (addFlag) v += *p;
                *p = v;
            }
        }
    }
}

// kpa path: tanh(kan(prod)) dotted with type-gated W -> Bsc[row]
__global__ void __launch_bounds__(128)
k_kan_pair(const float* __restrict__ Hst, const float* __restrict__ WcP,
           const float* __restrict__ W11, const float* __restrict__ W12,
           const float* __restrict__ W22, float* __restrict__ Bsc) {
    __shared__ float lds[5120];
    int row0 = blockIdx.x << 4;
    InPair in{Hst, row0};
    v8f acc = kan_gemm16<64, 64>(in, WcP, lds);
    int tid = threadIdx.x, wave = tid >> 5, lane = tid & 31, kh = lane >> 4;
    // (i, j-half) constant within a 16-row tile
    int i = (row0 >> 6) & 63, j = row0 & 63;
    const float* Wv = (i < 32) ? ((j < 32) ? W11 : W12) : ((j < 32) ? W12 : W22);
    float wv = Wv[(wave << 4) + (lane & 15)];
    if (tid < 16) lds[tid] = 0.0f;
    __syncthreads();
#pragma unroll
    for (int r = 0; r < 8; ++r) {
        float t = tanhf(acc[r]) * wv;
        for (int s = 1; s < 16; s <<= 1) t += __shfl_xor(t, s, 32);
        if ((lane & 15) == 0) atomicAdd(&lds[r + (kh << 3)], t);
    }
    __syncthreads();
    if (tid < 16) Bsc[row0 + tid] = lds[tid];
}

// stack-attention path: tanh(kan(h_st*S_pad)) @ W_m -> logits[row]
__global__ void __launch_bounds__(128)
k_kan_stack(const float* __restrict__ Hst, const float* __restrict__ S,
            const float* __restrict__ WcP, const float* __restrict__ Wm,
            float* __restrict__ logits) {
    __shared__ float lds[5120];
    int row0 = blockIdx.x << 4;
    InStack in{Hst, S, row0};
    v8f acc = kan_gemm16<64, 64>(in, WcP, lds);
    int tid = threadIdx.x, wave = tid >> 5, lane = tid & 31, kh = lane >> 4;
    float wv = Wm[(wave << 4) + (lane & 15)];
    if (tid < 16) lds[tid] = 0.0f;
    __syncthreads();
#pragma unroll
    for (int r = 0; r < 8; ++r) {
        float t = tanhf(acc[r]) * wv;
        for (int s = 1; s < 16; s <<= 1) t += __shfl_xor(t, s, 32);
        if ((lane & 15) == 0) atomicAdd(&lds[r + (kh << 3)], t);
    }
    __syncthreads();
    if (tid < 16) logits[row0 + tid] = lds[tid];
}

// row-wise softmax over 64 entries
__global__ void k_softmax64(const float* __restrict__ X, float* __restrict__ Y) {
    __shared__ float sm[64];
    int row = blockIdx.x, t = threadIdx.x;
    float v = X[(row << 6) + t];
    sm[t] = v; __syncthreads();
    for (int s = 32; s > 0; s >>= 1) { if (t < s) sm[t] = fmaxf(sm[t], sm[t + s]); __syncthreads(); }
    float mx = sm[0]; __syncthreads();
    float e = expf(v - mx);
    sm[t] = e; __syncthreads();
    for (int s = 32; s > 0; s >>= 1) { if (t < s) sm[t] += sm[t + s]; __syncthreads(); }
    Y[(row << 6) + t] = e / sm[0];
}

// h_w[b,f] = sum_n h_st[b,n,f] * A_m[b,n]
__global__ void k_hw(const float* __restrict__ Hst, const float* __restrict__ Am,
                     float* __restrict__ hw) {
    int b = blockIdx.x, f = threadIdx.x;
    float s = 0.0f;
    for (int n = 0; n < 64; ++n) s += Hst[(((b << 6) + n) << 6) + f] * Am[(b << 6) + n];
    hw[(b << 6) + f] = s;
}

// h_att[b] = B_hat[b] (64x64) @ h_st[b] (64x64), fp32 WMMA
__global__ void __launch_bounds__(128)
k_hatt(const float* __restrict__ Bh, const float* __restrict__ Hst,
       float* __restrict__ Hatt) {
    int b = blockIdx.x >> 2, i0 = (blockIdx.x & 3) << 4;
    int tid = threadIdx.x, wave = tid >> 5, lane = tid & 31;
    int m = lane & 15, kh = lane >> 4, n = lane & 15, c0 = wave << 4;
    const float* A = Bh + (((b << 6) + i0) << 6);
    const float* B = Hst + ((size_t)(b << 6) << 6);
    v8f acc = {};
#pragma unroll 4
    for (int kb = 0; kb < 64; kb += 4) {
        int k0 = kb + (kh << 1);
        v2f a = *(const v2f*)(A + (m << 6) + k0);   // contiguous pair, one b64
        v2f bb;
        bb[0] = B[(k0 << 6) + c0 + n];
        bb[1] = B[((k0 + 1) << 6) + c0 + n];
        acc = wmma4(a, bb, acc);
    }
#pragma unroll
    for (int r = 0; r < 8; ++r)
        Hatt[(((b << 6) + i0 + r + (kh << 3)) << 6) + c0 + n] = acc[r];
}

// BatchNorm stats over 512 rows x 64 channels
__global__ void k_bn_stats(const float* __restrict__ H, float* __restrict__ mu,
                           float* __restrict__ rsig) {
    int c = threadIdx.x;
    float s = 0.0f, s2 = 0.0f;
    for (int r = 0; r < 512; ++r) { float v = H[(r << 6) + c]; s += v; s2 += v * v; }
    float m = s * (1.0f / 512.0f);
    float var = s2 * (1.0f / 512.0f) - m * m;
    mu[c] = m;
    rsig[c] = rsqrtf(var + 1e-5f);
}

// normalize + split into out[:16384] (c<32) and out[16384:32768] (c>=32)
__global__ void k_bn_apply(const float* __restrict__ H, const float* __restrict__ mu,
                           const float* __restrict__ rsig, const float* __restrict__ g,
                           const float* __restrict__ be, float* __restrict__ out) {
    int idx = blockIdx.x * blockDim.x + threadIdx.x;
    if (idx < 32768) {
        int row = idx >> 6, c = idx & 63;
        float y = g[c] * (H[idx] - mu[c]) * rsig[c] + be[c];
        if (c < 32) out[(row << 5) + c] = y;
        else        out[16384 + (row << 5) + (c - 32)] = y;
    }
}

// ---------------- host ----------------
extern "C" void kernel_launch(void* const* d_in, const int* in_sizes, int n_in,
                              void* d_out, int out_size, void* d_ws, size_t ws_size,
                              hipStream_t stream) {
    (void)in_sizes; (void)n_in; (void)out_size; (void)ws_size;
#define IN(i) ((const float*)d_in[i])
    const float* h_t = IN(0);
    const float* h_s = IN(1);
    const float* Sst = IN(2);
    // params flattened in dict order: per-KAN {prelu_w, base_w, spline_w, coeffs}
    const int P_KT = 3, P_KS = 7, P_KPA = 11, P_KSA = 15, P_KSU1 = 19, P_KSU2 = 23,
              P_KHU1 = 27, P_KHU2 = 31;
    const float* W11 = IN(35); const float* W12 = IN(36); const float* W22 = IN(37);
    const float* Wm  = IN(38); const float* gam = IN(39); const float* bet = IN(40);

    float* ws = (float*)d_ws;
    float* Wc_kt   = ws;                  // 704*32
    float* Wc_ks   = Wc_kt   + 22528;
    float* Wc_kpa  = Wc_ks   + 22528;     // 1408*64
    float* Wc_ksa  = Wc_kpa  + 90112;
    float* Wc_ksu1 = Wc_ksa  + 90112;     // 1408*32
    float* Wc_ksu2 = Wc_ksu1 + 45056;     // 704*32
    float* Wc_khu1 = Wc_ksu2 + 22528;
    float* Wc_khu2 = Wc_khu1 + 90112;
    float* Hst     = Wc_khu2 + 90112;     // 512*64
    float* Bsc     = Hst  + 32768;        // 8*64*64
    float* Bhat    = Bsc  + 32768;
    float* logits  = Bhat + 32768;        // 512
    float* Am      = logits + 512;        // 512
    float* hw      = Am   + 512;          // 512
    float* Hatt    = hw   + 512;          // 512*64
    float* Hnew    = Hatt + 32768;
    float* mu      = Hnew + 32768;
    float* rsig    = mu + 64;

    float* out  = (float*)d_out;
    float* Snew = out + 32768;            // third output

    // build combined (pair-interleaved) weights
    auto prep = [&](int p, float* Wc, int fin, int fout) {
        int total = fin * 22 * fout;
        k_prep<<<(total + 255) / 256, 256, 0, stream>>>(
            IN(p), IN(p + 1), IN(p + 2), IN(p + 3), Wc, fin, fout);
    };
    prep(P_KT,   Wc_kt,   32, 32);
    prep(P_KS,   Wc_ks,   32, 32);
    prep(P_KPA,  Wc_kpa,  64, 64);
    prep(P_KSA,  Wc_ksa,  64, 64);
    prep(P_KSU1, Wc_ksu1, 64, 32);
    prep(P_KSU2, Wc_ksu2, 32, 32);
    prep(P_KHU1, Wc_khu1, 64, 64);
    prep(P_KHU2, Wc_khu2, 64, 64);

    // h_st = [kan(h_t; kt) | kan(h_s; ks)]
    k_kan_plain<32, 32><<<32, 128, 0, stream>>>(h_t, 32, Wc_kt, Hst, 64, 0,  0, 512);
    k_kan_plain<32, 32><<<32, 128, 0, stream>>>(h_s, 32, Wc_ks, Hst, 64, 32, 0, 512);

    // pairwise KAN + type-gated reduction -> Bsc, then softmax over j -> B_hat
    k_kan_pair<<<2048, 128, 0, stream>>>(Hst, Wc_kpa, W11, W12, W22, Bsc);
    k_softmax64<<<512, 64, 0, stream>>>(Bsc, Bhat);

    // stack attention -> A_m -> h_w
    k_kan_stack<<<32, 128, 0, stream>>>(Hst, Sst, Wc_ksa, Wm, logits);
    k_softmax64<<<8, 64, 0, stream>>>(logits, Am);
    k_hw<<<8, 64, 0, stream>>>(Hst, Am, hw);

    // S_new = kan(h_w; ksu1) + kan(S; ksu2)   (8 rows, padded to a 16-row tile)
    k_kan_plain<64, 32><<<1, 128, 0, stream>>>(hw,  64, Wc_ksu1, Snew, 32, 0, 0, 8);
    k_kan_plain<32, 32><<<1, 128, 0, stream>>>(Sst, 32, Wc_ksu2, Snew, 32, 0, 1, 8);

    // h_att = B_hat @ h_st ; h_new = kan(h_att; khu1) + kan(h_st; khu2)
    k_hatt<<<32, 128, 0, stream>>>(Bhat, Hst, Hatt);
    k_kan_plain<64, 64><<<32, 128, 0, stream>>>(Hatt, 64, Wc_khu1, Hnew, 64, 0, 0, 512);
    k_kan_plain<64, 64><<<32, 128, 0, stream>>>(Hst,  64, Wc_khu2, Hnew, 64, 0, 1, 512);

    // batchnorm (training-mode batch stats) + split outputs
    k_bn_stats<<<1, 64, 0, stream>>>(Hnew, mu, rsig);
    k_bn_apply<<<128, 256, 0, stream>>>(Hnew, mu, rsig, gam, bet, out);
#undef IN
}